// MultiTaskLoss_25675314495849
// MI455X (gfx1250) — compile-verified
//
#include <hip/hip_runtime.h>
#include <stdint.h>

// Problem constants (from the reference): N=8, C=19, H=512, W=1024
#define NIMG   8
#define NCLS   19
#define HW     (512 * 1024)        // 524288 pixels per image plane
#define PIX    (NIMG * HW)         // 4194304 total pixels
#define BLK    256                 // threads per block (8 wave32)
#define TILE_W 32                  // pixels per wave tile (1 per lane)
#define CHUNKS_PER_IMG (HW / BLK)  // 2048
#define CHUNKS (PIX / BLK)         // 16384
#define WS_STRIDE 40               // floats per block in workspace

typedef uint32_t v4u __attribute__((ext_vector_type(4)));
typedef int      v4i __attribute__((ext_vector_type(4)));
typedef int      v8i __attribute__((ext_vector_type(8)));

// ---- CDNA5 staging path selection ---------------------------------------
#if defined(__HIP_DEVICE_COMPILE__) && __has_builtin(__builtin_amdgcn_tensor_load_to_lds)
#define USE_TDM 1
#else
#define USE_TDM 0
#endif

#if defined(__HIP_DEVICE_COMPILE__)
#if USE_TDM
#define STAGE_PEND 1   // one TDM instruction outstanding per staged tile
#if __has_builtin(__builtin_amdgcn_s_wait_tensorcnt)
#define WAIT_STAGE(n) __builtin_amdgcn_s_wait_tensorcnt(n)
#else
#define WAIT_STAGE(n) asm volatile("s_wait_tensorcnt %0" ::"i"(n) : "memory")
#endif
#else
#define STAGE_PEND 19  // 19 async B32 loads outstanding per staged tile
#if __has_builtin(__builtin_amdgcn_s_wait_asynccnt)
#define WAIT_STAGE(n) __builtin_amdgcn_s_wait_asynccnt(n)
#else
#define WAIT_STAGE(n) asm volatile("s_wait_asynccnt %0" ::"i"(n) : "memory")
#endif
#endif
#define WAIT_DS0() asm volatile("s_wait_dscnt 0x0" ::: "memory")
#else
#define WAIT_STAGE(n) ((void)0)
#define WAIT_DS0() ((void)0)
#endif

// Generic pointer to an LDS object: low 32 bits are the LDS byte offset
// (shared aperture lives in addr[63:32] on CDNA5).
__device__ __forceinline__ uint32_t lds_byte_off(const void* p) {
  return (uint32_t)(uintptr_t)p;
}

#if USE_TDM
// One TDM descriptor: 2-D tile, 19 rows x 32 f32, row stride = HW elements.
// D# group0: [1:0]=count=1, [63:32]=lds_addr, [120:64]=global_addr, [127:126]=type=2
// D# group1: [17:16]=data_size=2 (4B), [79:48]=tensor_dim0=32, [111:80]=tensor_dim1=19,
//            [127:112]=tile_dim0=32, [143:128]=tile_dim1=19, [207:160]=dim0_stride=HW
__device__ __forceinline__ void tdm_load_tile(uint64_t ga, uint32_t lds_addr) {
  const v4u g0 = { 1u, lds_addr, (uint32_t)ga,
                   (uint32_t)(ga >> 32) | 0x80000000u };
  const v8i g1 = { 0x20000,                 // data_size = 4 bytes
                   (int)(TILE_W << 16),     // tensor_dim0[15:0] at bits[63:48]
                   (int)(NCLS << 16),       // dim0 hi=0 | tensor_dim1[15:0]
                   (int)(TILE_W << 16),     // dim1 hi=0 | tile_dim0
                   NCLS,                    // tile_dim1 | tile_dim2=0
                   (int)HW,                 // tensor_dim0_stride lo32
                   0, 0 };
  const v4i z4 = { 0, 0, 0, 0 };
#if __clang_major__ >= 23
  const v8i z8 = { 0, 0, 0, 0, 0, 0, 0, 0 };
  __builtin_amdgcn_tensor_load_to_lds(g0, g1, z4, z4, z8, 0);
#else
  __builtin_amdgcn_tensor_load_to_lds(g0, g1, z4, z4, 0);
#endif
}
#endif

// Stage one wave's 19x32 f32 tile for chunk k into its LDS region.
// Region layout: [c][32] floats, contiguous (matches TDM row-major packing).
__device__ __forceinline__ void stage_tile(const float* masks_pred, uint32_t k,
                                           uint32_t wv, uint32_t lane,
                                           uint32_t lds_region_byte) {
  const uint32_t n = k >> 11;                          // k / CHUNKS_PER_IMG
  const uint32_t r = (k & (CHUNKS_PER_IMG - 1)) << 8;  // pixel offset in plane
#if USE_TDM
  (void)lane;
  const uint64_t ga = (uint64_t)masks_pred +
      4ull * ((uint64_t)n * ((uint64_t)NCLS * HW) + r + wv * TILE_W);
  tdm_load_tile(ga, lds_region_byte);
#else
  const unsigned long long base = (unsigned long long)masks_pred;
#pragma unroll
  for (uint32_t c = 0; c < NCLS; ++c) {
    const uint32_t goff =
        (((n * (uint32_t)NCLS + c) * (uint32_t)HW) + r + wv * TILE_W + lane) * 4u;
    const uint32_t loff = lds_region_byte + (c * TILE_W + lane) * 4u;
#if defined(__HIP_DEVICE_COMPILE__)
    asm volatile("global_load_async_to_lds_b32 %0, %1, %2"
                 :: "v"(loff), "v"(goff), "s"(base)
                 : "memory");
#endif
  }
#endif
}

// ---- Stage 1: per-block partials (seg sums, counts, dep L1 sum) ---------
__global__ __launch_bounds__(BLK) void mtl_partials(
    const float* __restrict__ masks_pred,   // [N,C,H,W]
    const float* __restrict__ deps_pred,    // [N,1,H,W]
    const int*   __restrict__ true_masks,   // [N,H,W]
    const float* __restrict__ true_deps,    // [N,1,H,W]
    float* __restrict__ ws_part) {          // [gridDim.x][WS_STRIDE]
  __shared__ float s_tile[2][BLK / 32][NCLS * TILE_W];  // double-buffered tiles
  __shared__ float s_sum[NCLS];
  __shared__ float s_cnt[NCLS];
  __shared__ float s_red[BLK];

  const uint32_t tid = threadIdx.x;
  if (tid < NCLS) { s_sum[tid] = 0.0f; s_cnt[tid] = 0.0f; }
  __syncthreads();

  // wave-uniform ids so descriptor math stays on the scalar pipe
  const uint32_t wv   = ((uint32_t)__builtin_amdgcn_readfirstlane((int)tid)) >> 5;
  const uint32_t lane = tid & 31u;
  const uint32_t tile_bytes = NCLS * TILE_W * 4u;  // 2432 B per wave region
  const uint32_t lds0 = lds_byte_off(&s_tile[0][0][0]) + wv * tile_bytes;
  const uint32_t lds1 = lds_byte_off(&s_tile[1][0][0]) + wv * tile_bytes;

  float dep_acc = 0.0f;
  uint32_t k = blockIdx.x;
  const uint32_t stride = gridDim.x;
  int buf = 0;

  stage_tile(masks_pred, k, wv, lane, lds0);  // warm the per-wave pipeline

  for (;;) {
    const uint32_t kn = k + stride;
    const bool has_next = (kn < (uint32_t)CHUNKS);
    if (has_next) {
      WAIT_DS0();  // WAR guard: prior LDS reads of buf^1 retired before DMA write
      stage_tile(masks_pred, kn, wv, lane, buf ? lds0 : lds1);
      WAIT_STAGE(STAGE_PEND);  // in-order completion: current buffer landed
    } else {
      WAIT_STAGE(0);
    }

    const uint32_t n = k >> 11;
    const uint32_t r = (k & (CHUNKS_PER_IMG - 1)) << 8;
    const uint32_t p = n * (uint32_t)HW + r + tid;

    const float* region = &s_tile[buf][wv][0];
    float xs[NCLS];
#pragma unroll
    for (int c = 0; c < NCLS; ++c) xs[c] = region[c * TILE_W + lane];

    // argmax (strict > => first max, matching jnp.argmax) + running max
    float m = xs[0];
    int   am = 0;
#pragma unroll
    for (int c = 1; c < NCLS; ++c) {
      if (xs[c] > m) { m = xs[c]; am = c; }
    }

    const int t = true_masks[p];
    float xt = xs[0];
#pragma unroll
    for (int c = 1; c < NCLS; ++c) xt = (t == c) ? xs[c] : xt;

    float se = 0.0f;
#pragma unroll
    for (int c = 0; c < NCLS; ++c) se += __expf(xs[c] - m);

    const float lseg = m + __logf(se) - xt;  // = -log_softmax(x)[t]

    atomicAdd(&s_sum[am], lseg);             // ds_add_f32
    atomicAdd(&s_cnt[am], 1.0f);

    dep_acc += fabsf(deps_pred[p] - true_deps[p]);

    if (!has_next) break;
    k = kn;
    buf ^= 1;
  }

  // block-reduce the depth partial
  __syncthreads();
  s_red[tid] = dep_acc;
  __syncthreads();
#pragma unroll
  for (int off = BLK / 2; off > 0; off >>= 1) {
    if (tid < (uint32_t)off) s_red[tid] += s_red[tid + off];
    __syncthreads();
  }

  float* o = ws_part + (size_t)blockIdx.x * WS_STRIDE;
  if (tid < NCLS) {
    o[tid]        = s_sum[tid];
    o[NCLS + tid] = s_cnt[tid];
  }
  if (tid == 0) o[2 * NCLS] = s_red[0];
}

// ---- Stage 2: deterministic fixed-order combine -------------------------
__global__ __launch_bounds__(32) void mtl_finalize(
    const float* __restrict__ ws,       // [nB][WS_STRIDE]
    const float* __restrict__ weight,   // [NCLS]
    float* __restrict__ out, int nB) {
  __shared__ float contrib[32];
  const int c = threadIdx.x;

  float seg = 0.0f, cnt = 0.0f;
  if (c < NCLS) {
    for (int b = 0; b < nB; ++b) {
      seg += ws[(size_t)b * WS_STRIDE + c];
      cnt += ws[(size_t)b * WS_STRIDE + NCLS + c];
    }
  }
  const float pn = (cnt == 0.0f) ? 1.0f : cnt;
  contrib[c] = (c < NCLS) ? (weight[c] * seg) / pn : 0.0f;
  __syncthreads();

  if (c == 0) {
    float dep = 0.0f;
    for (int b = 0; b < nB; ++b) dep += ws[(size_t)b * WS_STRIDE + 2 * NCLS];
    float ls = 0.0f;
    for (int i = 0; i < NCLS; ++i) ls += contrib[i];
    ls *= (1.0f / (float)NCLS);              // mean over classes
    const float ld = dep / (float)PIX;       // mean over all pixels
    // begin == stop_gradient(current) on the first call -> x/x + y/y
    out[0] = ls / ls + ld / ld;
  }
}

extern "C" void kernel_launch(void* const* d_in, const int* in_sizes, int n_in,
                              void* d_out, int out_size, void* d_ws, size_t ws_size,
                              hipStream_t stream) {
  const float* loss_weight = (const float*)d_in[0];
  const float* masks_pred  = (const float*)d_in[1];
  const float* deps_pred   = (const float*)d_in[2];
  const int*   true_masks  = (const int*)d_in[3];
  const float* true_deps   = (const float*)d_in[4];
  float* out = (float*)d_out;
  float* ws  = (float*)d_ws;

  int nB = 1024;  // 16 chunks per block; saturates all WGPs
  const size_t per_blk = WS_STRIDE * sizeof(float);
  if (ws_size < (size_t)nB * per_blk) {
    nB = (int)(ws_size / per_blk);
    if (nB < 1) nB = 1;
  }
  if (nB > CHUNKS) nB = CHUNKS;

  mtl_partials<<<nB, BLK, 0, stream>>>(masks_pred, deps_pred, true_masks,
                                       true_deps, ws);
  mtl_finalize<<<1, 32, 0, stream>>>(ws, loss_weight, out, nB);
}